// Base2DPolygonRasterizer_64441689309405
// MI455X (gfx1250) — compile-verified
//
#include <hip/hip_runtime.h>
#include <math.h>

// ---------------------------------------------------------------------------
// Soft polygon rasterizer for MI455X (gfx1250, wave32).
//
// Roofline: ~67M pixel-segment interactions (~1.7 GFLOP) vs 4 MB output
// => compute bound. The segment-distance inner products are reformulated as
// rank-4 GEMMs so each 16-pixel x 16-segment tile is produced by two
// V_WMMA_F32_16X16X4_F32 ops (f32 WMMA is required: quadratic pixel features
// reach ~5e5, which overflows f16 and is too coarse in bf16):
//   |ap|^2 = [X^2+Y^2, X, Y, 1] . [1, -2ax, -2ay, ax^2+ay^2]
//   u      = [X,       Y, 1, 0] . [abx, aby, -(a.ab), 0]
//   d2     = |ap|^2 - t*(2u - t*denom),  t = clamp(u/denom, 0, 1)
// Coordinates are centered at (512,512) (exact fp32 shifts) to limit
// cancellation in the WMMA accumulation.
// ---------------------------------------------------------------------------

typedef __attribute__((ext_vector_type(2))) float v2f;
typedef __attribute__((ext_vector_type(8))) float v8f;

#define NSEG     64
#define IMG_W    1024
#define CENTER_F 512.0f
#define THRESH   30.0f

// Workspace layout (float offsets)
#define WS_F1  0     // F1[k][64], k=0..3 : {1, -2ax, -2ay, ax^2+ay^2}
#define WS_F2  256   // F2[k][64], k=0..3 : {abx, aby, -(a.ab), 0}
#define WS_DNM 512   // denom = max(|ab|^2, 1e-12)
#define WS_RDN 576   // 1/denom
#define WS_AX  640   // a.x (centered)
#define WS_AY  704   // a.y (centered)
#define WS_BX  768   // b.x (centered)
#define WS_SL  832   // aby / safe_abx
#define WS_BND 896   // {x_lo-T, x_hi+T, y_lo-T, y_hi+T} (uncentered bounds)

__global__ __launch_bounds__(64) void raster_prep(const float* __restrict__ poly,
                                                  float* __restrict__ ws) {
  int i = threadIdx.x;
  if (i < NSEG) {
    // OFFSET=(0,0), SCALE=1 fold away.
    float ax = poly[2 * i + 0] - CENTER_F;
    float ay = poly[2 * i + 1] - CENTER_F;
    int j = (i + 1) & (NSEG - 1);
    float bx = poly[2 * j + 0] - CENTER_F;
    float by = poly[2 * j + 1] - CENTER_F;
    float abx = bx - ax, aby = by - ay;
    float dnm = fmaxf(abx * abx + aby * aby, 1e-12f);

    ws[WS_F1 + 0 * 64 + i] = 1.0f;
    ws[WS_F1 + 1 * 64 + i] = -2.0f * ax;
    ws[WS_F1 + 2 * 64 + i] = -2.0f * ay;
    ws[WS_F1 + 3 * 64 + i] = ax * ax + ay * ay;

    ws[WS_F2 + 0 * 64 + i] = abx;
    ws[WS_F2 + 1 * 64 + i] = aby;
    ws[WS_F2 + 2 * 64 + i] = -(ax * abx + ay * aby);
    ws[WS_F2 + 3 * 64 + i] = 0.0f;

    ws[WS_DNM + i] = dnm;
    ws[WS_RDN + i] = 1.0f / dnm;
    ws[WS_AX + i] = ax;
    ws[WS_AY + i] = ay;
    ws[WS_BX + i] = bx;
    float safe_abx = (abx == 0.0f) ? 1.0f : abx;
    ws[WS_SL + i] = aby / safe_abx;
  }
  if (i == 0) {
    float mnx = 1e30f, mxx = -1e30f, mny = 1e30f, mxy = -1e30f;
    for (int k = 0; k < NSEG; ++k) {
      float x = poly[2 * k + 0], y = poly[2 * k + 1];
      mnx = fminf(mnx, x); mxx = fmaxf(mxx, x);
      mny = fminf(mny, y); mxy = fmaxf(mxy, y);
    }
    ws[WS_BND + 0] = floorf(mnx) - THRESH;
    ws[WS_BND + 1] = floorf(mxx) + 1.0f + THRESH;
    ws[WS_BND + 2] = floorf(mny) - THRESH;
    ws[WS_BND + 3] = floorf(mxy) + 1.0f + THRESH;
  }
}

// One wave32 = one 16-pixel row strip vs all 64 segments (4 tiles of 16).
__global__ __launch_bounds__(256) void raster_main(const float* __restrict__ ws,
                                                   float* __restrict__ out) {
  const int lane = threadIdx.x & 31;
  const int tile = blockIdx.x * 8 + (threadIdx.x >> 5);
  const int row = tile >> 6;         // 0..1023
  const int x0 = (tile & 63) << 4;   // strip start (multiple of 16)
  const int half = lane >> 4;        // lane half (0/1)
  const int nIdx = lane & 15;

  const float Yc = (float)row - CENTER_F;
  const float Xa = (float)(x0 + nIdx) - CENTER_F;  // A-matrix pixel M = lane&15

  // A layout (16x4 f32): VGPR0 = K(half*2), VGPR1 = K(half*2+1), M = lane&15.
  v2f A1, A2;
  A1.x = half ? Yc : (Xa * Xa + Yc * Yc);  // K0 = X^2+Y^2 | K2 = Y
  A1.y = half ? 1.0f : Xa;                 // K1 = X       | K3 = 1
  A2.x = half ? 1.0f : Xa;                 // K0 = X       | K2 = 1
  A2.y = half ? 0.0f : Yc;                 // K1 = Y       | K3 = 0

  const int kLo = half ? 2 : 0;
  const int kHi = half ? 3 : 1;

  float d2min[8];
  int par[8];
#pragma unroll
  for (int v = 0; v < 8; ++v) { d2min[v] = 1e30f; par[v] = 0; }

#pragma unroll
  for (int t = 0; t < 4; ++t) {
    const int s = t * 16 + nIdx;  // this lane's segment column N = lane&15

    v2f B1, B2;
    B1.x = ws[WS_F1 + kLo * 64 + s];
    B1.y = ws[WS_F1 + kHi * 64 + s];
    B2.x = ws[WS_F2 + kLo * 64 + s];
    B2.y = ws[WS_F2 + kHi * 64 + s];

    const float dnm = ws[WS_DNM + s];
    const float rdn = ws[WS_RDN + s];
    const float sax = ws[WS_AX + s];
    const float say = ws[WS_AY + s];
    const float sbx = ws[WS_BX + s];
    const float slp = ws[WS_SL + s];

    v8f zero = {0.f, 0.f, 0.f, 0.f, 0.f, 0.f, 0.f, 0.f};
    // D = A x B : 16 pixels x 16 segments, K=4
    v8f AP2 = __builtin_amdgcn_wmma_f32_16x16x4_f32(
        false, A1, false, B1, (short)0, zero, false, false);
    v8f U = __builtin_amdgcn_wmma_f32_16x16x4_f32(
        false, A2, false, B2, (short)0, zero, false, false);

#pragma unroll
    for (int v = 0; v < 8; ++v) {
      // C layout: element (M = v + 8*half, N = lane&15)
      const float Xc = (float)(x0 + v + 8 * half) - CENTER_F;
      const float u = U[v];
      const float ap2 = AP2[v];
      const float tt = fminf(fmaxf(u * rdn, 0.0f), 1.0f);
      const float d2 = ap2 - tt * (2.0f * u - tt * dnm);
      d2min[v] = fminf(d2min[v], d2);

      const bool crosses = (sax <= Xc) != (sbx <= Xc);
      const float yint = fmaf(Xc - sax, slp, say);
      par[v] ^= (crosses && (yint <= Yc)) ? 1 : 0;
    }
  }

  // Reduce over the 16 segment lanes within each half (wave32 xor shuffles).
#pragma unroll
  for (int v = 0; v < 8; ++v) {
#pragma unroll
    for (int m = 1; m < 16; m <<= 1) {
      d2min[v] = fminf(d2min[v], __shfl_xor(d2min[v], m, 32));
      par[v] ^= __shfl_xor(par[v], m, 32);
    }
  }

  // Lane j (< 8) of each half writes pixel x0 + j + 8*half.
  const int j = lane & 15;
  float dm = d2min[0];
  int pr = par[0];
#pragma unroll
  for (int v = 1; v < 8; ++v) {
    if (j == v) { dm = d2min[v]; pr = par[v]; }
  }

  if (j < 8) {
    const int px = x0 + j + 8 * half;
    const int py = row;
    const float pxf = (float)px, pyf = (float)py;
    const float bxlo = ws[WS_BND + 0], bxhi = ws[WS_BND + 1];
    const float bylo = ws[WS_BND + 2], byhi = ws[WS_BND + 3];

    const bool inside = (pr & 1) != 0;
    const bool in_band =
        (pxf >= bxlo) && (pxf <= bxhi) && (pyf >= bylo) && (pyf <= byhi);
    const bool zero_far = (!inside) && (dm >= THRESH * THRESH);

    const float z = inside ? dm : -dm;       // indicator * d2 / SIGMA
    const float val = 1.0f / (1.0f + __expf(-z));
    out[py * IMG_W + px] = (in_band && !zero_far) ? val : 0.0f;
  }
}

extern "C" void kernel_launch(void* const* d_in, const int* in_sizes, int n_in,
                              void* d_out, int out_size, void* d_ws, size_t ws_size,
                              hipStream_t stream) {
  (void)in_sizes; (void)n_in; (void)out_size; (void)ws_size;
  const float* poly = (const float*)d_in[0];
  float* ws = (float*)d_ws;
  float* out = (float*)d_out;

  raster_prep<<<1, 64, 0, stream>>>(poly, ws);
  // 1024 rows * 64 strips = 65536 waves; 8 waves (256 threads) per block.
  raster_main<<<8192, 256, 0, stream>>>(ws, out);
}